// LossRelDepth_58514634440839
// MI455X (gfx1250) — compile-verified
//
#include <hip/hip_runtime.h>
#include <hip/hip_bf16.h>

#define BSZ 512
#define LM 68
#define IMG 224
#define RS 7
#define R2 49
#define LL (LM * LM)          // 4624
#define NTILE 5               // ceil(68/16)
#define TPB (NTILE * NTILE)   // 25 tiles per batch
#define NWAVE (BSZ * TPB)     // 12800 waves in main pass
#define NBLK3 (NWAVE / 8)     // 1600 blocks, 8 waves each
#define MASK_THR 0.18f        // VALID_RANGE / DEPTH_SCALE

typedef __attribute__((ext_vector_type(2))) float v2f;
typedef __attribute__((ext_vector_type(8))) float v8f;

// ---------------------------------------------------------------------------
// Kernel 1: per-region median via grid-sample + rank-count order statistic.
// 4 regions per 256-thread block; 64 threads per region (49 active).
// ---------------------------------------------------------------------------
__global__ void k_region_median(const float* __restrict__ depth,
                                const float* __restrict__ landmarks,
                                const float* __restrict__ scalef,
                                const float* __restrict__ bbox,
                                float* __restrict__ med) {
  __shared__ float sv[4][64];
  __shared__ int scnt[4];
  const int tid = threadIdx.x;
  const int r = tid >> 6;
  const int t = tid & 63;
  const int region = blockIdx.x * 4 + r;
  const int b = region / LM;
  const int l = region % LM;
  if (t == 0) scnt[r] = 0;
  float v = 0.0f;
  if (t < R2) {
    const int oi = t / RS;   // y-offset index
    const int oj = t % RS;   // x-offset index
    const float lmx = landmarks[(size_t)(b * LM + l) * 2 + 0];
    const float lmy = landmarks[(size_t)(b * LM + l) * 2 + 1];
    const float bx = bbox[b * 4 + 0], by = bbox[b * 4 + 1];
    const float sx = scalef[b * 2 + 0], sy = scalef[b * 2 + 1];
    // face_lm = (lm - bbox) * scale / IMG * 2 - 1   (mirror op order)
    const float flx = ((lmx - bx) * sx) / (float)IMG * 2.0f - 1.0f;
    const float fly = ((lmy - by) * sy) / (float)IMG * 2.0f - 1.0f;
    // x = linspace(-3.5, 3.5, 7) / IMG * 2
    const float step = 7.0f / 6.0f;
    const float offx = (-3.5f + (float)oj * step) / (float)IMG * 2.0f;
    const float offy = (-3.5f + (float)oi * step) / (float)IMG * 2.0f;
    const float gx = flx + offx;
    const float gy = fly + offy;
    const float ix = ((gx + 1.0f) * (float)IMG - 1.0f) * 0.5f;
    const float iy = ((gy + 1.0f) * (float)IMG - 1.0f) * 0.5f;
    int pxi = (int)rintf(ix);   // round-half-even, matches jnp.round
    int pyi = (int)rintf(iy);
    const bool valid = (pxi >= 0) && (pxi < IMG) && (pyi >= 0) && (pyi < IMG);
    pxi = min(max(pxi, 0), IMG - 1);
    pyi = min(max(pyi, 0), IMG - 1);
    v = depth[(size_t)b * IMG * IMG + (size_t)pyi * IMG + pxi] * (valid ? 1.0f : 0.0f);
    sv[r][t] = v;
  }
  __syncthreads();
  if (t < R2 && v <= 1e-4f) atomicAdd(&scnt[r], 1);
  __syncthreads();
  if (t < R2) {
    const int cnt = scnt[r];
    const int st = min(max(cnt, 1), R2 - 1);
    const int k = (st + R2 - 1) >> 1;       // median index in sorted order
    int lt = 0, eq = 0;
    for (int j = 0; j < R2; ++j) {
      const float w = sv[r][j];
      lt += (w < v) ? 1 : 0;
      eq += ((w == v) && (j < t)) ? 1 : 0;
    }
    if (lt + eq == k) med[region] = v;      // exactly one lane matches
  }
}

// ---------------------------------------------------------------------------
// Kernel 2: per-batch lower median of 68 medians (order statistic k=33).
// ---------------------------------------------------------------------------
__global__ void k_mom(const float* __restrict__ med, float* __restrict__ mom) {
  __shared__ float sm[LM];
  const int b = blockIdx.x;
  const int i = threadIdx.x;
  if (i < LM) sm[i] = med[b * LM + i];
  __syncthreads();
  if (i < LM) {
    const float v = sm[i];
    int lt = 0, eq = 0;
    for (int j = 0; j < LM; ++j) {
      const float w = sm[j];
      lt += (w < v) ? 1 : 0;
      eq += ((w == v) && (j < i)) ? 1 : 0;
    }
    if (lt + eq == (LM - 1) / 2) mom[b] = v;
  }
}

// ---------------------------------------------------------------------------
// Kernel 3: main bandwidth pass. One wave per 16x16 tile of (i,j).
// diff tile computed with V_WMMA_F32_16X16X4_F32 as a K=2 rank-2 update:
//   D[m][n] = A[m,0]*B[0,n] + A[m,1]*B[1,n] = (500*m_i)*1 + 1*(-500*m_j)
// Mask bits carried in wave-uniform ballot masks (no LDS permutes).
// ---------------------------------------------------------------------------
__global__ void k_loss_tiles(const float* __restrict__ rel,
                             const float* __restrict__ med,
                             const float* __restrict__ mom,
                             float* __restrict__ out_diff,
                             float* __restrict__ out_mask,
                             float* __restrict__ part) {
  __shared__ float wls[8], wms[8];
  const int lane = threadIdx.x & 31;
  const int wv = threadIdx.x >> 5;
  const int w = blockIdx.x * 8 + wv;
  const int b = w / TPB;
  const int t = w % TPB;
  const int i0 = (t / NTILE) * 16;
  const int j0 = (t % NTILE) * 16;
  const float momv = mom[b];

  // lanes 0..15 hold row/col medians for this tile (clamped unconditional load)
  const int li = i0 + lane;
  const int lj = j0 + lane;
  const float rmRaw = med[b * LM + min(li, LM - 1)];
  const float cmRaw = med[b * LM + min(lj, LM - 1)];
  const bool rOk = (lane < 16) && (li < LM);
  const bool cOk = (lane < 16) && (lj < LM);
  const float rowMed = rOk ? rmRaw : 0.0f;
  const float colMed = cOk ? cmRaw : 0.0f;
  // validity masks as wave-uniform bitmasks (bit M / bit N)
  const unsigned int rowMask =
      (unsigned int)__ballot(rOk && (fabsf(rmRaw - momv) < MASK_THR));
  const unsigned int colMask =
      (unsigned int)__ballot(cOk && (fabsf(cmRaw - momv) < MASK_THR));

  // A (16x4): lanes 0-15 -> K=0,1 ; lanes 16-31 -> K=2,3 (zeroed)
  v2f av, bv;
  av.x = (lane < 16) ? rowMed * 500.0f : 0.0f;   // K=0: 500*m_i
  av.y = (lane < 16) ? 1.0f : 0.0f;              // K=1: 1
  // B (4x16): VGPR0 lanes 0-15 = K=0 row ; VGPR1 lanes 0-15 = K=1 row
  bv.x = (lane < 16) ? 1.0f : 0.0f;              // K=0: ones
  bv.y = (lane < 16) ? -(colMed * 500.0f) : 0.0f; // K=1: -500*m_j
  v8f cz = {};
  v8f d = __builtin_amdgcn_wmma_f32_16x16x4_f32(false, av, false, bv,
                                                (short)0, cz, false, false);

  const int N = lane & 15;
  const int Mbase = (lane >> 4) << 3;            // D layout: lanes>=16 are M=r+8
  const unsigned int colBit = (colMask >> N) & 1u;
  const size_t base = (size_t)b * LL;
  float ls = 0.0f, ms = 0.0f;

  if ((i0 + 16 <= LM) && (j0 + 16 <= LM)) {
    // ---- full tile fast path: batched loads, no guards ----
    float rv[8];
#pragma unroll
    for (int r = 0; r < 8; ++r) {
      const size_t idx = base + (size_t)(i0 + Mbase + r) * LM + (j0 + N);
      rv[r] = __builtin_nontemporal_load(&rel[idx]);
    }
#pragma unroll
    for (int r = 0; r < 8; ++r) {
      const size_t idx = base + (size_t)(i0 + Mbase + r) * LM + (j0 + N);
      const float diff = d[r];
      const float fp = (float)(((rowMask >> (Mbase + r)) & colBit) & 1u);
      const float dd = rv[r] - diff;
      const float ad = fabsf(dd);
      const float le = (ad < 1.0f) ? (0.5f * dd * dd) : (ad - 0.5f);
      ls += le * fp;
      ms += fp;
      __builtin_nontemporal_store(diff, &out_diff[idx]);
      __builtin_nontemporal_store(fp, &out_mask[idx]);
    }
  } else {
    // ---- edge tile: per-element guards ----
#pragma unroll
    for (int r = 0; r < 8; ++r) {
      const int gi = i0 + Mbase + r;
      const int gj = j0 + N;
      if (gi < LM && gj < LM) {
        const size_t idx = base + (size_t)gi * LM + gj;
        const float diff = d[r];
        const float fp = (float)(((rowMask >> (Mbase + r)) & colBit) & 1u);
        const float rvv = __builtin_nontemporal_load(&rel[idx]);
        const float dd = rvv - diff;
        const float ad = fabsf(dd);
        const float le = (ad < 1.0f) ? (0.5f * dd * dd) : (ad - 0.5f);
        ls += le * fp;
        ms += fp;
        __builtin_nontemporal_store(diff, &out_diff[idx]);
        __builtin_nontemporal_store(fp, &out_mask[idx]);
      }
    }
  }

  // deterministic wave reduction (fixed xor-tree)
#pragma unroll
  for (int off = 16; off >= 1; off >>= 1) {
    ls += __shfl_xor(ls, off, 32);
    ms += __shfl_xor(ms, off, 32);
  }
  if (lane == 0) { wls[wv] = ls; wms[wv] = ms; }
  __syncthreads();
  if (threadIdx.x == 0) {
    float bl = 0.0f, bm = 0.0f;
    for (int k = 0; k < 8; ++k) { bl += wls[k]; bm += wms[k]; }  // fixed order
    part[2 * blockIdx.x + 0] = bl;
    part[2 * blockIdx.x + 1] = bm;
  }
}

// ---------------------------------------------------------------------------
// Kernel 4: deterministic final reduction over 1600 block partials.
// ---------------------------------------------------------------------------
__global__ void k_finalize(const float* __restrict__ part, float* __restrict__ out) {
  __shared__ float sl[256], sm[256];
  const int tid = threadIdx.x;
  float ls = 0.0f, ms = 0.0f;
  for (int i = tid; i < NBLK3; i += 256) {   // fixed strided order
    ls += part[2 * i + 0];
    ms += part[2 * i + 1];
  }
  sl[tid] = ls; sm[tid] = ms;
  __syncthreads();
  for (int s = 128; s > 0; s >>= 1) {        // fixed tree
    if (tid < s) { sl[tid] += sl[tid + s]; sm[tid] += sm[tid + s]; }
    __syncthreads();
  }
  if (tid == 0) out[0] = sl[0] / (sm[0] + 1e-4f);
}

// ---------------------------------------------------------------------------
extern "C" void kernel_launch(void* const* d_in, const int* in_sizes, int n_in,
                              void* d_out, int out_size, void* d_ws, size_t ws_size,
                              hipStream_t stream) {
  const float* rel    = (const float*)d_in[0];   // [512,68,68]
  const float* depth  = (const float*)d_in[1];   // [512,1,224,224]
  const float* lmk    = (const float*)d_in[2];   // [512,68,2]
  const float* scalef = (const float*)d_in[3];   // [512,2]
  const float* bbox   = (const float*)d_in[4];   // [512,4]

  float* out = (float*)d_out;                    // loss | diff | mask
  float* out_loss = out;
  float* out_diff = out + 1;
  float* out_mask = out + 1 + (size_t)BSZ * LL;

  float* ws      = (float*)d_ws;
  float* ws_med  = ws;                           // BSZ*LM   = 34816 floats
  float* ws_mom  = ws + BSZ * LM;                // BSZ      = 512 floats
  float* ws_part = ws + BSZ * LM + BSZ;          // 2*NBLK3  = 3200 floats

  k_region_median<<<(BSZ * LM) / 4, 256, 0, stream>>>(depth, lmk, scalef, bbox, ws_med);
  k_mom<<<BSZ, 128, 0, stream>>>(ws_med, ws_mom);
  k_loss_tiles<<<NBLK3, 256, 0, stream>>>(rel, ws_med, ws_mom, out_diff, out_mask, ws_part);
  k_finalize<<<1, 256, 0, stream>>>(ws_part, out_loss);
}